// GCN_59742995087372
// MI455X (gfx1250) — compile-verified
//
#include <hip/hip_runtime.h>

#define DIM 64
#define WPAD 68   // padded row length for transposed W in LDS (bank-conflict-free b64 reads)

typedef __attribute__((ext_vector_type(2))) float v2f;
typedef __attribute__((ext_vector_type(8))) float v8f;

// ---------------------------------------------------------------------------
// Degree / normalization precompute
// ---------------------------------------------------------------------------
__global__ void gcn_init_deg(float* __restrict__ deg, int n) {
    int i = blockIdx.x * blockDim.x + threadIdx.x;
    if (i < n) deg[i] = 1.0f;  // self loop
}

__global__ void gcn_count_deg(const int* __restrict__ dst, float* __restrict__ deg, int e) {
    int i = blockIdx.x * blockDim.x + threadIdx.x;
    if (i < e) unsafeAtomicAdd(&deg[dst[i]], 1.0f);
}

__global__ void gcn_make_dinv(float* __restrict__ d, int n) {
    int i = blockIdx.x * blockDim.x + threadIdx.x;
    if (i < n) d[i] = rsqrtf(d[i]);
}

// ---------------------------------------------------------------------------
// H = relu?(X) @ W   (X: n x 64 row-major, W: 64 x 64 row-major, H: n x 64)
// One wave computes a 16-row x 64-col tile with V_WMMA_F32_16X16X4_F32.
// Block = 256 threads = 8 waves = 128 rows. W staged TRANSPOSED in LDS so a
// lane's B fragment (W[k][n], W[k+1][n]) is one aligned ds_load_b64.
// ReLU of the previous layer is fused into the X-tile staging.
// ---------------------------------------------------------------------------
__global__ __launch_bounds__(256) void gcn_gemm_wmma(const float* __restrict__ X,
                                                     const float* __restrict__ W,
                                                     float* __restrict__ H, int n,
                                                     int relu) {
    __shared__ float sWt[DIM * WPAD];    // ~17 KB, sWt[nCol*WPAD + k] = W[k][nCol]
    __shared__ float sX[8][16 * DIM];    // 32 KB (16x64 tile per wave)

    const int tid  = threadIdx.x;
    const int wave = tid >> 5;
    const int lane = tid & 31;

    // stage W transposed into LDS (coalesced global reads)
    for (int i = tid; i < DIM * DIM; i += 256) {
        int k = i >> 6;
        int c = i & 63;
        sWt[c * WPAD + k] = W[i];
    }

    const long rowBase = ((long)blockIdx.x * 8 + wave) * 16;

    // stage this wave's 16x64 X tile into LDS as float4, fusing previous ReLU
    float* sx = &sX[wave][0];
    #pragma unroll
    for (int j = 0; j < 8; ++j) {
        int idx = j * 32 + lane;                 // float4 index 0..255 within tile
        long r  = rowBase + (idx >> 4);          // 16 float4 per row
        if (r >= n) r = n - 1;                   // clamp; stores guarded later
        float4 v = ((const float4*)X)[r * 16 + (idx & 15)];
        if (relu) {
            v.x = fmaxf(v.x, 0.0f); v.y = fmaxf(v.y, 0.0f);
            v.z = fmaxf(v.z, 0.0f); v.w = fmaxf(v.w, 0.0f);
        }
        ((float4*)sx)[idx] = v;
    }
    __syncthreads();

    const int half = lane >> 4;        // 0: K pair {0,1}; 1: K pair {2,3}
    const int sub  = lane & 15;        // A: row M, B/C: col N
    const int koff = half * 2;

    v8f c0 = {}, c1 = {}, c2 = {}, c3 = {};

    #pragma unroll
    for (int k0 = 0; k0 < DIM; k0 += 4) {
        v2f a  = *(const v2f*)&sx[sub * DIM + k0 + koff];          // A[M][k],A[M][k+1]
        v2f b0 = *(const v2f*)&sWt[(sub +  0) * WPAD + k0 + koff]; // W[k][n],W[k+1][n]
        v2f b1 = *(const v2f*)&sWt[(sub + 16) * WPAD + k0 + koff];
        v2f b2 = *(const v2f*)&sWt[(sub + 32) * WPAD + k0 + koff];
        v2f b3 = *(const v2f*)&sWt[(sub + 48) * WPAD + k0 + koff];

        c0 = __builtin_amdgcn_wmma_f32_16x16x4_f32(false, a, false, b0, (short)0, c0, false, false);
        c1 = __builtin_amdgcn_wmma_f32_16x16x4_f32(false, a, false, b1, (short)0, c1, false, false);
        c2 = __builtin_amdgcn_wmma_f32_16x16x4_f32(false, a, false, b2, (short)0, c2, false, false);
        c3 = __builtin_amdgcn_wmma_f32_16x16x4_f32(false, a, false, b3, (short)0, c3, false, false);
    }

    // C layout: VGPR i, lanes 0-15 -> M=i, lanes 16-31 -> M=i+8; N = lane&15
    #pragma unroll
    for (int i = 0; i < 8; ++i) {
        long r = rowBase + i + 8 * half;
        if (r < n) {
            float* out = H + r * DIM + sub;
            out[ 0] = c0[i];
            out[16] = c1[i];
            out[32] = c2[i];
            out[48] = c3[i];
        }
    }
}

// ---------------------------------------------------------------------------
// acc[i,d] = h[i,d] * dinv[i]^2 + b[d]     (self-loop term + bias)
// ---------------------------------------------------------------------------
__global__ void gcn_init_acc(const float* __restrict__ h, const float* __restrict__ dinv,
                             const float* __restrict__ b, float* __restrict__ acc, int n) {
    int t = blockIdx.x * blockDim.x + threadIdx.x;
    if (t < n * DIM) {
        int i = t >> 6;
        int d = t & 63;
        float w = dinv[i];
        acc[t] = h[t] * w * w + b[d];
    }
}

// ---------------------------------------------------------------------------
// acc[dst] += dinv[src]*dinv[dst] * h[src]; 16 threads per edge, float4 each.
// Working set (~64MB) is L2-resident; unsafeAtomicAdd -> global_atomic_add_f32.
// ---------------------------------------------------------------------------
__global__ void gcn_scatter(const float* __restrict__ h, const int* __restrict__ src,
                            const int* __restrict__ dst, const float* __restrict__ dinv,
                            float* __restrict__ acc, int e) {
    long t = (long)blockIdx.x * blockDim.x + threadIdx.x;
    long ed = t >> 4;
    if (ed >= e) return;
    int part = (int)(t & 15);
    int s = src[ed];
    int d = dst[ed];
    float w = dinv[s] * dinv[d];
    float4 hv = ((const float4*)h)[(long)s * 16 + part];
    float* out = acc + (long)d * DIM + part * 4;
    unsafeAtomicAdd(out + 0, w * hv.x);
    unsafeAtomicAdd(out + 1, w * hv.y);
    unsafeAtomicAdd(out + 2, w * hv.z);
    unsafeAtomicAdd(out + 3, w * hv.w);
}

// ---------------------------------------------------------------------------
// Host-side orchestration
// ---------------------------------------------------------------------------
extern "C" void kernel_launch(void* const* d_in, const int* in_sizes, int n_in,
                              void* d_out, int out_size, void* d_ws, size_t ws_size,
                              hipStream_t stream) {
    const int*   edge = (const int*)d_in[0];
    const float* emb  = (const float*)d_in[1];
    const float* Ws[3] = { (const float*)d_in[2], (const float*)d_in[4], (const float*)d_in[6] };
    const float* Bs[3] = { (const float*)d_in[3], (const float*)d_in[5], (const float*)d_in[7] };

    const int E = in_sizes[0] / 2;
    const int n = in_sizes[1] / DIM;
    const int* src = edge;
    const int* dst = edge + E;

    char* ws = (char*)d_ws;
    const size_t featBytes = (size_t)n * DIM * sizeof(float);
    float* dinv = (float*)ws;                          // n floats
    float* h    = (float*)(ws + 524288);               // n x 64
    float* acc  = (float*)(ws + 524288 + featBytes);   // n x 64

    const int B = 256;
    dim3 gN((n + B - 1) / B);
    dim3 gE((E + B - 1) / B);
    dim3 gF((n * DIM + B - 1) / B);
    dim3 gS((int)(((long)E * 16 + B - 1) / B));
    dim3 gG((n + 127) / 128);

    // normalization: dinv = rsqrt(deg), deg = 1 + indegree
    gcn_init_deg <<<gN, B, 0, stream>>>(dinv, n);
    gcn_count_deg<<<gE, B, 0, stream>>>(dst, dinv, E);
    gcn_make_dinv<<<gN, B, 0, stream>>>(dinv, n);

    const float* x = emb;
    for (int l = 0; l < 3; ++l) {
        // layers 1,2 read previous acc with ReLU fused into tile staging
        gcn_gemm_wmma<<<gG, B, 0, stream>>>(x, Ws[l], h, n, l > 0 ? 1 : 0);
        float* a = (l == 2) ? (float*)d_out : acc;
        gcn_init_acc<<<gF, B, 0, stream>>>(h, dinv, Bs[l], a, n);
        gcn_scatter <<<gS, B, 0, stream>>>(h, src, dst, dinv, a, E);
        x = acc;
    }
}